// VIN_28286654611801
// MI455X (gfx1250) — compile-verified
//
#include <hip/hip_runtime.h>
#include <hip/hip_bf16.h>
#include <math.h>

// ---------------------------------------------------------------------------
// VIN on gfx1250: one workgroup per image; r and v live in LDS for the whole
// 37-step value-iteration chain; each VI step = implicit GEMM done with
// V_WMMA_F32_16X16X4_F32 (full fp32 precision, matches reference).
// D = W(16ch x K) * Data(K x 16px)  ->  channel-major D: max over channels is
// per-lane VALU + one v_permlanex16 half-swap (no LDS round-trip).
// ---------------------------------------------------------------------------

typedef float v2f __attribute__((ext_vector_type(2)));
typedef float v8f __attribute__((ext_vector_type(8)));

#define NB   128
#define HH   64
#define WW   64
#define CHH  150
#define CHQ  10
#define PAD  66            // 64 + 1 halo each side (zero padding)

// LDS layout (floats)
#define OFF_RPAD   0                    // 66*66            = 4356
#define OFF_R2     4356                 // union region
//   phase A view of region2:
#define R2_W0      0                    // 2700
#define R2_WR      2700                 // 1350
#define R2_B0      4050                 // 150
//   phase B view of region2:
#define R2_VA      0                    // 4356
#define R2_VB      4356                 // 4356
#define R2_WMQ     8712                 // 20*16 = 320 (padded weight matrix)
#define R2_WMSW    9032                 // 320
#define R2_WMSW2   9352                 // 320
#define OFF_QSEL   (4356 + 9672)        // 10 (+pad)
#define SMEM_FLOATS (4356 + 9672 + 16)  // 14044 floats = 56176 B < 64 KB

// Output flat layout: logits[128*8] | probs[128*8] | q[128*64*64*10] | q_out[128*10]
#define OUT_PROBS  1024
#define OUT_Q      2048
#define OUT_QOUT   (2048 + 128*64*64*10)

// xor-16 half swap + max, as pure VALU (v_permlanex16_b32 with identity
// selects: lane i reads lane i of the opposite 16-lane group).
__device__ __forceinline__ float halfswap_max(float m) {
  int mi = __float_as_int(m);
  int sw = __builtin_amdgcn_permlanex16(mi, mi, 0x76543210, (int)0xfedcba98,
                                        false, false);
  return fmaxf(m, __int_as_float(sw));
}

// Load A operand (the 16x20-padded weight matrix) once per step.
// A is MxK (16 channels x 4): lane ln holds rows K=k0..k0+1 (lower half) /
// k0+2..k0+3 (upper half) for channel M=ln.
__device__ __forceinline__ void load_W(const float* __restrict__ wmat,
                                       int ln, int half, v2f* W)
{
  #pragma unroll
  for (int kc = 0; kc < 5; ++kc) {
    int k = kc * 4 + half * 2;
    v2f w;
    w.x = wmat[k * 16 + ln];
    w.y = wmat[(k + 1) * 16 + ln];
    W[kc] = w;
  }
}

// 5 chained fp32 WMMAs: D(16ch x 16px) += W(16x4) * Data(4x16).
// B operand per chunk: lane ln = pixel column; even K = r plane, odd K = v
// plane at spatial tap s. K=18,19 rows of W are zero, so the clamped (s=8)
// data loaded for them contributes exactly 0 — no divergent guard needed.
__device__ __forceinline__ void vi_conv_tile(
    const float* __restrict__ rpad, const float* __restrict__ vcur,
    const v2f* __restrict__ W, int y, int xb, int ln, int half, v8f& C)
{
  #pragma unroll
  for (int kc = 0; kc < 5; ++kc) {
    int s  = kc * 2 + half;              // spatial tap 0..9
    int sc = (s > 8) ? 8 : s;            // clamp: tap 9 pairs with zero W rows
    int dy = sc / 3, dx = sc - dy * 3;
    int addr = (y + dy) * PAD + (xb + ln + dx);
    v2f Bd;
    Bd.x = rpad[addr];                   // even K = r plane
    Bd.y = vcur[addr];                   // odd  K = v plane
    C = __builtin_amdgcn_wmma_f32_16x16x4_f32(
            false, W[kc], false, Bd, (short)0, C, false, false);
  }
}

__device__ __forceinline__ void vi_step(
    const float* __restrict__ rpad, const float* __restrict__ vcur,
    float* __restrict__ vnext, const float* __restrict__ wmat,
    int ln, int half)
{
  v2f W[5];
  load_W(wmat, ln, half, W);
  const int wave = threadIdx.x >> 5;
  const int xb   = (wave & 3) << 4;      // loop-invariant column block
  for (int y = wave >> 2; y < HH; y += 2) {
    v8f C = {0.f,0.f,0.f,0.f,0.f,0.f,0.f,0.f};
    vi_conv_tile(rpad, vcur, W, y, xb, ln, half, C);
    // v[pixel] = max over 10 channels. Channel-major D: lower-half lane ln
    // holds ch 0-7 of pixel ln in C[0..7]; upper-half lane ln holds ch 8-9
    // in C[0..1]. One VALU half-swap combines them.
    float m01 = fmaxf(C[0], C[1]);
    float m8  = fmaxf(fmaxf(m01, fmaxf(C[2], C[3])),
                      fmaxf(fmaxf(C[4], C[5]), fmaxf(C[6], C[7])));
    float m = half ? m01 : m8;
    m = halfswap_max(m);
    if (half == 0)
      vnext[(y + 1) * PAD + (xb + ln + 1)] = m;   // 16 consecutive lanes
  }
}

__device__ __forceinline__ void vi_final(
    const float* __restrict__ rpad, const float* __restrict__ vcur,
    const float* __restrict__ wmat, float* __restrict__ qimg,
    int s1, int s2, float* __restrict__ qsel, int ln, int half)
{
  v2f W[5];
  load_W(wmat, ln, half, W);
  const int wave = threadIdx.x >> 5;
  const int xb   = (wave & 3) << 4;
  for (int y = wave >> 2; y < HH; y += 2) {
    v8f C = {0.f,0.f,0.f,0.f,0.f,0.f,0.f,0.f};
    vi_conv_tile(rpad, vcur, W, y, xb, ln, half, C);
    // Channel-major D: lane ln owns pixel xb+ln; lower half stores ch 0-7
    // contiguously (4x b64), upper half stores ch 8-9 (1x b64).
    int px = xb + ln;
    float* qp = qimg + (y * WW + px) * CHQ + half * 8;
    float2 p01; p01.x = C[0]; p01.y = C[1];
    if (half == 0) {
      float2 p23; p23.x = C[2]; p23.y = C[3];
      float2 p45; p45.x = C[4]; p45.y = C[5];
      float2 p67; p67.x = C[6]; p67.y = C[7];
      ((float2*)qp)[0] = p01;
      ((float2*)qp)[1] = p23;
      ((float2*)qp)[2] = p45;
      ((float2*)qp)[3] = p67;
    } else {
      ((float2*)qp)[0] = p01;              // channels 8,9
    }
    if (y == s1 && px == s2) {
      if (half == 0) {
        #pragma unroll
        for (int j = 0; j < 8; ++j) qsel[j] = C[j];
      } else {
        qsel[8] = C[0]; qsel[9] = C[1];
      }
    }
  }
}

__global__ __launch_bounds__(256)
void vin_kernel(const float* __restrict__ x,
                const int*   __restrict__ S1,
                const int*   __restrict__ S2,
                const int*   __restrict__ VInum,
                const float* __restrict__ w0,
                const float* __restrict__ b0,
                const float* __restrict__ w_r,
                const float* __restrict__ w_q,
                const float* __restrict__ w_sw,
                const float* __restrict__ w_sw2,
                const float* __restrict__ w_dense,
                float*       __restrict__ out)
{
  __shared__ float smem[SMEM_FLOATS];
  const int img  = blockIdx.x;
  const int tid  = threadIdx.x;
  const int lane = tid & 31;
  const int ln   = lane & 15;
  const int half = lane >> 4;

  float* rpad = smem + OFF_RPAD;
  float* r2   = smem + OFF_R2;
  float* qsel = smem + OFF_QSEL;

  const float* xim = x + (size_t)img * (HH * WW * 2);
  __builtin_prefetch(xim + tid * 32, 0, 1);     // pull x image toward L2/L0

  // ---- Stage 1: zero r_pad, stage weights into LDS, build padded W tables
  for (int i = tid; i < PAD * PAD; i += 256) rpad[i] = 0.f;
  for (int i = tid; i < 2700; i += 256) r2[R2_W0 + i] = w0[i];
  for (int i = tid; i < 1350; i += 256) r2[R2_WR + i] = w_r[i];
  for (int i = tid; i < 150;  i += 256) r2[R2_B0 + i] = b0[i];
  for (int i = tid; i < 320; i += 256) {        // 20(K-pad) x 16(ch-pad)
    int k = i >> 4, n = i & 15;
    float vq = 0.f, vs = 0.f, vs2 = 0.f;
    if (k < 18 && n < CHQ) {
      vq  = w_q  [k * CHQ + n];
      vs  = w_sw [k * CHQ + n];
      vs2 = w_sw2[k * CHQ + n];
    }
    r2[R2_WMQ + i] = vq; r2[R2_WMSW + i] = vs; r2[R2_WMSW2 + i] = vs2;
  }
  __syncthreads();

  // ---- Stage 2: fused conv0 (+bias) and conv_r -> r_pad (scatter + LDS atomics)
  {
    const float* w0l = r2 + R2_W0;
    const float* wrl = r2 + R2_WR;
    const float* b0l = r2 + R2_B0;
    for (int p = tid; p < HH * WW; p += 256) {
      int Y = p >> 6, X = p & 63;
      float xv[18];
      #pragma unroll
      for (int ky = 0; ky < 3; ++ky) {
        #pragma unroll
        for (int kx = 0; kx < 3; ++kx) {
          int yy = Y + ky - 1, xx = X + kx - 1;
          int j = (ky * 3 + kx) * 2;
          float a0 = 0.f, a1 = 0.f;
          if (yy >= 0 && yy < HH && xx >= 0 && xx < WW) {
            const float* xp = xim + (yy * WW + xx) * 2;
            a0 = xp[0]; a1 = xp[1];
          }
          xv[j] = a0; xv[j + 1] = a1;
        }
      }
      float t[9];
      #pragma unroll
      for (int s = 0; s < 9; ++s) t[s] = 0.f;
      for (int c = 0; c < CHH; ++c) {
        float hc = b0l[c];
        #pragma unroll
        for (int j = 0; j < 18; ++j) hc = fmaf(xv[j], w0l[j * CHH + c], hc);
        #pragma unroll
        for (int s = 0; s < 9; ++s) t[s] = fmaf(hc, wrl[s * CHH + c], t[s]);
      }
      // h[Y][X] contributes to r[Y-ky+1][X-kx+1] with weight w_r[ky][kx][:]
      #pragma unroll
      for (int ky = 0; ky < 3; ++ky) {
        #pragma unroll
        for (int kx = 0; kx < 3; ++kx) {
          int ry = Y + 1 - ky, rx = X + 1 - kx;
          if (ry >= 0 && ry < HH && rx >= 0 && rx < WW)
            atomicAdd(&rpad[(ry + 1) * PAD + (rx + 1)], t[ky * 3 + kx]);
        }
      }
    }
  }
  __syncthreads();

  // ---- Stage 3: zero v ping-pong buffers (overwrites phase-A weights)
  float* va = r2 + R2_VA;
  float* vb = r2 + R2_VB;
  for (int i = tid; i < 2 * PAD * PAD; i += 256) va[i] = 0.f;
  __syncthreads();

  // ---- Stage 4: value iteration with WMMA f32 16x16x4
  const float* wmq   = r2 + R2_WMQ;
  const float* wmsw  = r2 + R2_WMSW;
  const float* wmsw2 = r2 + R2_WMSW2;

  vi_step(rpad, va, vb, wmq, ln, half);          // first step, w_q weights
  __syncthreads();

  float* bufs[2] = { va, vb };
  int cur = 1;                                   // vb holds current v
  int km1 = VInum[0] - 1;                        // k-1 shared-weight steps
  for (int it = 0; it < km1; ++it) {
    vi_step(rpad, bufs[cur], bufs[1 - cur], wmsw, ln, half);
    __syncthreads();
    cur = 1 - cur;
  }

  // ---- Stage 5: final step (w_sw2): write q + gather at (S1,S2)
  const int s1 = S1[img];
  const int s2 = S2[img];
  float* qimg = out + OUT_Q + (size_t)img * (HH * WW * CHQ);
  vi_final(rpad, bufs[cur], wmsw2, qimg, s1, s2, qsel, ln, half);
  __syncthreads();

  // ---- Stage 6: q_out, dense 10->8, softmax
  if (tid < CHQ) out[OUT_QOUT + img * CHQ + tid] = qsel[tid];
  if (tid == 0) {
    float lg[8], ex[8];
    float mx = -3.0e38f;
    #pragma unroll
    for (int a = 0; a < 8; ++a) {
      float acc = 0.f;
      #pragma unroll
      for (int c = 0; c < CHQ; ++c) acc = fmaf(qsel[c], w_dense[c * 8 + a], acc);
      lg[a] = acc; mx = fmaxf(mx, acc);
    }
    float sum = 0.f;
    #pragma unroll
    for (int a = 0; a < 8; ++a) { ex[a] = __expf(lg[a] - mx); sum += ex[a]; }
    float inv = 1.f / sum;
    #pragma unroll
    for (int a = 0; a < 8; ++a) {
      out[img * 8 + a]             = lg[a];
      out[OUT_PROBS + img * 8 + a] = ex[a] * inv;
    }
  }
}

extern "C" void kernel_launch(void* const* d_in, const int* in_sizes, int n_in,
                              void* d_out, int out_size, void* d_ws, size_t ws_size,
                              hipStream_t stream) {
  (void)in_sizes; (void)n_in; (void)out_size; (void)d_ws; (void)ws_size;
  vin_kernel<<<NB, 256, 0, stream>>>(
      (const float*)d_in[0],   // x
      (const int*)  d_in[1],   // S1
      (const int*)  d_in[2],   // S2
      (const int*)  d_in[3],   // VInum
      (const float*)d_in[4],   // w0
      (const float*)d_in[5],   // b0
      (const float*)d_in[6],   // w_r
      (const float*)d_in[7],   // w_q
      (const float*)d_in[8],   // w_sw
      (const float*)d_in[9],   // w_sw2
      (const float*)d_in[10],  // w_dense
      (float*)d_out);
}